// RRWPLinearEdgeEncoder_59596966199890
// MI455X (gfx1250) — compile-verified
//
#include <hip/hip_runtime.h>

typedef __attribute__((ext_vector_type(2))) float v2f;
typedef __attribute__((ext_vector_type(8))) float v8f;

#define KSTEPS 21
#define KPAD   24
#define EMB    64
#define EPB    128   // edges per block in the WMMA kernel
#define THREADS 256

// ---------------- kernel 1: full-graph edge index (row-major, graph contiguous) ----
__global__ void k_init_idx(float* __restrict__ out, const int* __restrict__ n_ptr,
                           long long P) {
    long long p = (long long)blockIdx.x * blockDim.x + threadIdx.x;
    if (p >= P) return;
    long long n  = (long long)n_ptr[0];
    long long nn = n * n;
    long long g   = p / nn;
    long long rem = p - g * nn;
    long long row = g * n + rem / n;
    long long col = g * n + rem % n;
    out[p]     = (float)row;
    out[P + p] = (float)col;
}

// ---------------- kernel 2: zero out_val with wide stores ------------------------
__global__ void k_zero(float4* __restrict__ v, long long count4) {
    long long i = (long long)blockIdx.x * blockDim.x + threadIdx.x;
    if (i < count4) v[i] = make_float4(0.f, 0.f, 0.f, 0.f);
}

// ---------------- kernel 3: scatter-add edge_attr --------------------------------
// 32-bit element offsets off the uniform base -> saddr-form global_atomic_add_f32.
__global__ void k_edge_scatter(const float* __restrict__ ea,
                               const long long* __restrict__ eidx,
                               float* __restrict__ outv,
                               const int* __restrict__ n_ptr, int E) {
    long long gid = (long long)blockIdx.x * blockDim.x + threadIdx.x;
    if (gid >= (long long)E * EMB) return;
    int e = (int)(gid >> 6);
    int c = (int)(gid & 63);
    long long n = (long long)n_ptr[0];
    long long u = eidx[e];
    long long v = eidx[(long long)E + e];
    long long g = u / n;
    int pid = (int)(g * n * n + (u - g * n) * n + (v - g * n));
    unsigned idx = (unsigned)pid * (unsigned)EMB + (unsigned)c;  // < 2^31 elements
    unsafeAtomicAdd(outv + idx, ea[gid]);
}

// ---------------- kernel 4: fused WMMA projection + scatter-add ------------------
// Each block: 256 threads = 8 waves, 128 rrwp edges.
// Each wave: 16 edges -> D = A(16x24) x B(24x64) via V_WMMA_F32_16X16X4_F32,
// then fp32 atomic scatter of the 16x64 tile into out_val.
__global__ void k_rrwp_wmma(const float* __restrict__ rrwp,
                            const float* __restrict__ fcw,
                            const long long* __restrict__ ridx,
                            float* __restrict__ outv,
                            const int* __restrict__ n_ptr, int E) {
    __shared__ float aLDS[EPB * KPAD];   // 12 KB: 128 rrwp rows, K padded to 24
    __shared__ float bLDS[EMB * KPAD];   // 6 KB : fc_weight rows (B[k][n] = fcw[n*21+k])
    __shared__ int   pidLDS[EPB];

    const int tid  = threadIdx.x;
    const int base = blockIdx.x * EPB;
    const long long n = (long long)n_ptr[0];
    const bool fullBlock = (base + EPB) <= E;   // uniform across block

    // stage A rows (coalesced) with zero pad K=21..23
    for (int i = tid; i < EPB * KPAD; i += THREADS) {
        int r = i / KPAD, k = i - r * KPAD;
        int e = base + r;
        float val = 0.f;
        if (k < KSTEPS && e < E) val = rrwp[(long long)e * KSTEPS + k];
        aLDS[i] = val;
    }
    // stage B = fc_weight (n-major, stride KPAD), zero padded
    for (int i = tid; i < EMB * KPAD; i += THREADS) {
        int nn = i / KPAD, k = i - nn * KPAD;
        bLDS[i] = (k < KSTEPS) ? fcw[nn * KSTEPS + k] : 0.f;
    }
    // per-edge destination pair ids
    if (tid < EPB) {
        int e = base + tid;
        int pid = -1;
        if (e < E) {
            long long u = ridx[e];
            long long v = ridx[(long long)E + e];
            long long g = u / n;
            pid = (int)(g * n * n + (u - g * n) * n + (v - g * n));
        }
        pidLDS[tid] = pid;
    }
    __syncthreads();

    const int wave  = tid >> 5;       // 0..7
    const int lane  = tid & 31;
    const int lhalf = lane >> 4;      // 0/1
    const int l16   = lane & 15;
    const int kb    = lhalf * 2;      // K sub-pair owned by this half-wave

    // A fragments: lane holds row M=l16 of this wave's tile, K = 4c+kb, 4c+kb+1
    const float* Arow = &aLDS[(wave * 16 + l16) * KPAD];
    v2f afrag[6];
#pragma unroll
    for (int c = 0; c < 6; ++c) {
        afrag[c].x = Arow[4 * c + kb];
        afrag[c].y = Arow[4 * c + kb + 1];
    }

    // Preload this lane's 8 destination row ids ONCE (rows M = 8*lhalf + r).
    // Destinations as 32-bit element offsets off the uniform `outv` base:
    // out_val is 2^26 floats (<2^31), so saddr-form atomics apply and the
    // +t*16 term folds into the instruction immediate offset.
    int pidreg[8];
#pragma unroll
    for (int r = 0; r < 8; ++r)
        pidreg[r] = pidLDS[wave * 16 + 8 * lhalf + r];

    unsigned rowoff[8];
#pragma unroll
    for (int r = 0; r < 8; ++r)
        rowoff[r] = (unsigned)pidreg[r] * (unsigned)EMB + (unsigned)l16;

#pragma unroll
    for (int t = 0; t < 4; ++t) {     // 4 tiles of 16 along EMB
        const float* Brow = &bLDS[(t * 16 + l16) * KPAD];
        v8f acc = {0.f, 0.f, 0.f, 0.f, 0.f, 0.f, 0.f, 0.f};
#pragma unroll
        for (int c = 0; c < 6; ++c) { // K = 24 in chunks of 4
            v2f b;
            b.x = Brow[4 * c + kb];
            b.y = Brow[4 * c + kb + 1];
            acc = __builtin_amdgcn_wmma_f32_16x16x4_f32(
                /*neg_a=*/false, afrag[c], /*neg_b=*/false, b,
                /*c_mod=*/(short)0, acc, /*reuse_a=*/false, /*reuse_b=*/false);
        }
        // scatter: lane owns column N=t*16+l16 for rows M = r + 8*lhalf.
        // Fast path (uniform): all 128 edges valid -> unguarded hardware atomics,
        // EXEC stays all-ones, no per-element saveexec churn.
        if (fullBlock) {
#pragma unroll
            for (int r = 0; r < 8; ++r)
                unsafeAtomicAdd(outv + (rowoff[r] + t * 16), acc[r]);
        } else {
#pragma unroll
            for (int r = 0; r < 8; ++r)
                if (pidreg[r] >= 0)
                    unsafeAtomicAdd(outv + (rowoff[r] + t * 16), acc[r]);
        }
    }
}

// ---------------- launcher -------------------------------------------------------
extern "C" void kernel_launch(void* const* d_in, const int* in_sizes, int n_in,
                              void* d_out, int out_size, void* d_ws, size_t ws_size,
                              hipStream_t stream) {
    (void)n_in; (void)d_ws; (void)ws_size;

    const float*     rrwp  = (const float*)d_in[0];
    const float*     ea    = (const float*)d_in[1];
    const float*     fcw   = (const float*)d_in[2];
    const long long* ridx  = (const long long*)d_in[3];
    const long long* eidx  = (const long long*)d_in[4];
    const int*       n_ptr = (const int*)d_in[5];

    const int E_rrwp = in_sizes[3] / 2;
    const int E_real = in_sizes[4] / 2;

    // out = [row(P) | col(P) | val(P*64)]  =>  P = out_size / 66
    const long long P = (long long)out_size / (2 + EMB);

    float* out  = (float*)d_out;
    float* outv = out + 2 * P;

    // 1) indices
    {
        long long nb = (P + THREADS - 1) / THREADS;
        k_init_idx<<<(unsigned)nb, THREADS, 0, stream>>>(out, n_ptr, P);
    }
    // 2) zero out_val (float4 stores)
    {
        long long c4 = P * (EMB / 4);
        long long nb = (c4 + THREADS - 1) / THREADS;
        k_zero<<<(unsigned)nb, THREADS, 0, stream>>>((float4*)outv, c4);
    }
    // 3) edge_attr scatter
    {
        long long work = (long long)E_real * EMB;
        long long nb   = (work + THREADS - 1) / THREADS;
        k_edge_scatter<<<(unsigned)nb, THREADS, 0, stream>>>(ea, eidx, outv, n_ptr, E_real);
    }
    // 4) fused projection (WMMA) + scatter
    {
        long long nb = ((long long)E_rrwp + EPB - 1) / EPB;
        k_rrwp_wmma<<<(unsigned)nb, THREADS, 0, stream>>>(rrwp, fcw, ridx, outv, n_ptr, E_rrwp);
    }
}